// PFNet7_17781164606149
// MI455X (gfx1250) — compile-verified
//
#include <hip/hip_runtime.h>
#include <hip/hip_bf16.h>
#include <math.h>

// ---------------------------------------------------------------------------
// PFNet7 forward for MI455X (gfx1250, wave32).
// - Distance GEMM (16384^2 x 8) + top-40 fused with V_WMMA_F32_16X16X4_F32,
//   s/sq L2-resident, top-k lists in LDS.
// - MLP heads (64|82 ->126->126->126->6) as 16-node-batched f32 WMMA GEMMs:
//   cuts weight L2 traffic 16x (4.1 GB -> 0.26 GB) and moves ~2.8 GFLOP onto
//   the matrix pipe.
// - Small per-node kernels batched 4 nodes/block to amortize weight fetches.
// ---------------------------------------------------------------------------

typedef __attribute__((ext_vector_type(2))) float v2f;
typedef __attribute__((ext_vector_type(8))) float v8f;

#define KNN 40
#define INF_F 3.0e38f

__device__ __forceinline__ float lrelu(float a) { return a >= 0.0f ? a : 0.5f * a; }

// ---------------------------------------------------------------------------
// Kernel 1: nn1 encoder (12->64->64->12, LeakyReLU 0.5) + lin_s (->8) +
// lin_h (->22) + |s|^2.  4 nodes per 256-thread block (weights hit WGP$).
// ---------------------------------------------------------------------------
__global__ __launch_bounds__(256) void k_encoder(
    const float* __restrict__ x0,
    const float* __restrict__ W0, const float* __restrict__ b0,
    const float* __restrict__ W1, const float* __restrict__ b1,
    const float* __restrict__ W2, const float* __restrict__ b2,
    const float* __restrict__ Ws, const float* __restrict__ bs,
    const float* __restrict__ Wh, const float* __restrict__ bh,
    float* __restrict__ x, float* __restrict__ s, float* __restrict__ h,
    float* __restrict__ sq)
{
    __shared__ float in12[4][12], a1[4][64], a2[4][64], xo[4][12], so[4][8];
    const int sub = threadIdx.x >> 6, tt = threadIdx.x & 63;
    const int n = blockIdx.x * 4 + sub;
    if (tt < 12) in12[sub][tt] = x0[n * 12 + tt];
    __syncthreads();
    {   // 12 -> 64
        float acc = b0[tt];
#pragma unroll
        for (int k = 0; k < 12; ++k) acc += in12[sub][k] * W0[k * 64 + tt];
        a1[sub][tt] = lrelu(acc);
    }
    __syncthreads();
    {   // 64 -> 64
        float acc = b1[tt];
        for (int k = 0; k < 64; ++k) acc += a1[sub][k] * W1[k * 64 + tt];
        a2[sub][tt] = lrelu(acc);
    }
    __syncthreads();
    if (tt < 12) {   // 64 -> 12
        float acc = b2[tt];
        for (int k = 0; k < 64; ++k) acc += a2[sub][k] * W2[k * 12 + tt];
        float v = lrelu(acc);
        xo[sub][tt] = v;
        x[n * 12 + tt] = v;
    }
    __syncthreads();
    if (tt < 8) {    // lin_s: 12 -> 8 (no activation)
        float acc = bs[tt];
#pragma unroll
        for (int k = 0; k < 12; ++k) acc += xo[sub][k] * Ws[k * 8 + tt];
        so[sub][tt] = acc;
        s[n * 8 + tt] = acc;
    }
    if (tt < 22) {   // lin_h: 12 -> 22 (no activation)
        float acc = bh[tt];
#pragma unroll
        for (int k = 0; k < 12; ++k) acc += xo[sub][k] * Wh[k * 22 + tt];
        h[n * 22 + tt] = acc;
    }
    __syncthreads();
    if (tt == 0) {
        float acc = 0.0f;
#pragma unroll
        for (int k = 0; k < 8; ++k) acc += so[sub][k] * so[sub][k];
        sq[n] = acc;
    }
}

// ---------------------------------------------------------------------------
// Kernel 2: fused distance + top-40 (unchanged from round 0; WMMA confirmed).
// ---------------------------------------------------------------------------
#define TK_ROWS 16
#define TK_WAVES 8

__global__ __launch_bounds__(256) void k_dist_topk(
    const float* __restrict__ s, const float* __restrict__ sq,
    float* __restrict__ d2top, int* __restrict__ idxtop, int N)
{
    __shared__ float tile[TK_WAVES][16][17];          // padded: stride 17
    __shared__ float ld[TK_WAVES][TK_ROWS][KNN];      // sorted ascending d2
    __shared__ int   li[TK_WAVES][TK_ROWS][KNN];
    __shared__ float rowS[TK_ROWS][8];
    __shared__ float rowQ[TK_ROWS];

    const int tid = threadIdx.x;
    const int w = tid >> 5;
    const int l = tid & 31;
    const int rowBase = blockIdx.x * TK_ROWS;

    for (int i = tid; i < TK_WAVES * TK_ROWS * KNN; i += 256) {
        ((float*)ld)[i] = INF_F;
        ((int*)li)[i] = 0;
    }
    if (tid < TK_ROWS * 8) ((float*)rowS)[tid] = s[rowBase * 8 + tid];
    if (tid < TK_ROWS) rowQ[tid] = sq[rowBase + tid];
    __syncthreads();

    // A fragment (16x4 f32): lanes 0-15: M=lane, K={0,1}; lanes 16-31:
    // M=lane-16, K={2,3}.  -2 folded in so WMMA yields -2*si.sj.
    const int hr = l & 15;
    const int koff = (l < 16) ? 0 : 2;
    v2f a0, a1;
    a0.x = -2.0f * rowS[hr][koff];
    a0.y = -2.0f * rowS[hr][koff + 1];
    a1.x = -2.0f * rowS[hr][4 + koff];
    a1.y = -2.0f * rowS[hr][4 + koff + 1];

    const int ntiles = N >> 4;
    for (int ct = w; ct < ntiles; ct += TK_WAVES) {
        const int colBase = ct << 4;
        const int c = colBase + hr;
        v2f b0, b1;
        b0.x = s[c * 8 + koff];
        b0.y = s[c * 8 + koff + 1];
        b1.x = s[c * 8 + 4 + koff];
        b1.y = s[c * 8 + 4 + koff + 1];

        v8f acc = {0.f, 0.f, 0.f, 0.f, 0.f, 0.f, 0.f, 0.f};
        acc = __builtin_amdgcn_wmma_f32_16x16x4_f32(
            false, a0, false, b0, (short)0, acc, false, false);
        acc = __builtin_amdgcn_wmma_f32_16x16x4_f32(
            false, a1, false, b1, (short)0, acc, false, false);

        const float cq = sq[c];
        const int mo = (l < 16) ? 0 : 8;
#pragma unroll
        for (int r = 0; r < 8; ++r) {
            const int M = r + mo;
            float v = acc[r] + rowQ[M] + cq;
            if (rowBase + M == c) v = INF_F;
            tile[w][M][hr] = v;
        }
        __builtin_amdgcn_wave_barrier();

        if (l < 16) {
            const int m = l;
            float worst = ld[w][m][KNN - 1];
            for (int n2 = 0; n2 < 16; ++n2) {
                const float v = tile[w][m][n2];
                if (v < worst) {
                    int p = KNN - 1;
                    while (p > 0 && ld[w][m][p - 1] > v) {
                        ld[w][m][p] = ld[w][m][p - 1];
                        li[w][m][p] = li[w][m][p - 1];
                        --p;
                    }
                    ld[w][m][p] = v;
                    li[w][m][p] = colBase + n2;
                    worst = ld[w][m][KNN - 1];
                }
            }
        }
        __builtin_amdgcn_wave_barrier();
    }
    __syncthreads();

    if (w == 0 && l < TK_ROWS) {
        const int m = l;
        const int node = rowBase + m;
        int p[TK_WAVES];
#pragma unroll
        for (int q = 0; q < TK_WAVES; ++q) p[q] = 0;
        for (int o = 0; o < KNN; ++o) {
            float best = INF_F;
            int bw = 0;
#pragma unroll
            for (int q = 0; q < TK_WAVES; ++q) {
                const float v = (p[q] < KNN) ? ld[q][m][p[q]] : INF_F;
                if (v < best) { best = v; bw = q; }
            }
            d2top[node * KNN + o] = best;
            idxtop[node * KNN + o] = li[bw][m][p[bw]];
#pragma unroll
            for (int q = 0; q < TK_WAVES; ++q)
                if (q == bw) p[q]++;
        }
    }
}

// ---------------------------------------------------------------------------
// Kernel 3: GravNet aggregation + lin_out 56->64.  4 nodes per block.
// ---------------------------------------------------------------------------
__global__ __launch_bounds__(256) void k_gravnet_agg(
    const float* __restrict__ x, const float* __restrict__ h,
    const float* __restrict__ d2top, const int* __restrict__ idxtop,
    const float* __restrict__ Wo, const float* __restrict__ bo,
    float* __restrict__ x2)
{
    __shared__ float wL[4][KNN];
    __shared__ int iL[4][KNN];
    __shared__ float agg[4][44];
    __shared__ float xL[4][12];
    const int sub = threadIdx.x >> 6, tt = threadIdx.x & 63;
    const int n = blockIdx.x * 4 + sub;
    if (tt < KNN) {
        wL[sub][tt] = expf(-10.0f * d2top[n * KNN + tt]);
        iL[sub][tt] = idxtop[n * KNN + tt];
    }
    if (tt >= KNN && tt < KNN + 12) xL[sub][tt - KNN] = x[n * 12 + (tt - KNN)];
    __syncthreads();
    if (tt < 22) {
        float mean = 0.0f, mx = -INF_F;
        for (int k = 0; k < KNN; ++k) {
            const float v = h[iL[sub][k] * 22 + tt] * wL[sub][k];
            mean += v;
            mx = fmaxf(mx, v);
        }
        agg[sub][tt] = mean * (1.0f / (float)KNN);
        agg[sub][22 + tt] = mx;
    }
    __syncthreads();
    {   // lin_out: concat(agg[44], x[12]) @ Wo[56,64] + bo
        float o = bo[tt];
        for (int k = 0; k < 44; ++k) o += agg[sub][k] * Wo[k * 64 + tt];
#pragma unroll
        for (int k = 0; k < 12; ++k) o += xL[sub][k] * Wo[(44 + k) * 64 + tt];
        x2[n * 64 + tt] = o;
    }
}

// ---------------------------------------------------------------------------
// Kernel 4: GraphConv.  4 nodes per block.
// ---------------------------------------------------------------------------
__global__ __launch_bounds__(256) void k_graphconv(
    const float* __restrict__ x2,
    const float* __restrict__ d2top, const int* __restrict__ idxtop,
    const float* __restrict__ Wrel, const float* __restrict__ brel,
    const float* __restrict__ Wroot,
    float* __restrict__ x3)
{
    __shared__ float wL[4][KNN];
    __shared__ int iL[4][KNN];
    __shared__ float nbrL[4][64];
    __shared__ float xsL[4][64];
    const int sub = threadIdx.x >> 6, tt = threadIdx.x & 63;
    const int n = blockIdx.x * 4 + sub;
    if (tt < KNN) {
        wL[sub][tt] = expf(-10.0f * d2top[n * KNN + tt]);
        iL[sub][tt] = idxtop[n * KNN + tt];
    }
    __syncthreads();
    float nb = 0.0f;
    for (int k = 0; k < KNN; ++k) nb += wL[sub][k] * x2[iL[sub][k] * 64 + tt];
    nbrL[sub][tt] = nb;
    xsL[sub][tt] = x2[n * 64 + tt];
    __syncthreads();
    float o = brel[tt];
    for (int k = 0; k < 64; ++k)
        o += nbrL[sub][k] * Wrel[k * 64 + tt] + xsL[sub][k] * Wroot[k * 64 + tt];
    x3[n * 64 + tt] = o;
}

// ---------------------------------------------------------------------------
// WMMA MLP head: 16 nodes/block, each layer is M=16 x N=126 x K GEMM on
// V_WMMA_F32_16X16X4_F32.  N padded to 128 (8 waves = 8 N-tiles); padded
// columns compute exactly 0 (B and bias zeroed) so stores are unguarded and
// double as the K-padding zeros for the next layer.
// Activation pitches 100/132 (== 4 mod 32 dwords) avoid LDS bank conflicts
// on the A-fragment ds_load_b64s.
// ---------------------------------------------------------------------------
#define PIN 100    // input activation pitch (K up to 96)
#define PH 132     // hidden activation pitch (K up to 128)

__device__ __forceinline__ void layer_wmma(
    const float* __restrict__ actIn, int pIn,
    float* __restrict__ actOut, int pOut,
    const float* __restrict__ W, const float* __restrict__ bias,
    int dinL, int Kpad, int Nout, int w, int l)
{
    const int ntc = (Nout + 15) >> 4;
    if (w < ntc) {
        const int hr = l & 15;
        const int koff = (l < 16) ? 0 : 2;
        const int n = w * 16 + hr;
        const bool nok = (n < Nout);
        v8f acc = {0.f, 0.f, 0.f, 0.f, 0.f, 0.f, 0.f, 0.f};
        for (int k0 = 0; k0 < Kpad; k0 += 4) {
            const int ka = k0 + koff;
            v2f a, b;
            a.x = actIn[hr * pIn + ka];
            a.y = actIn[hr * pIn + ka + 1];
            b.x = (nok && ka < dinL) ? W[ka * Nout + n] : 0.0f;
            b.y = (nok && ka + 1 < dinL) ? W[(ka + 1) * Nout + n] : 0.0f;
            acc = __builtin_amdgcn_wmma_f32_16x16x4_f32(
                false, a, false, b, (short)0, acc, false, false);
        }
        const float bn = nok ? bias[n] : 0.0f;
        const int mo = (l < 16) ? 0 : 8;
#pragma unroll
        for (int r = 0; r < 8; ++r)
            actOut[(r + mo) * pOut + n] = lrelu(acc[r] + bn);
    }
}

__global__ __launch_bounds__(256) void k_head_wmma(
    const float* __restrict__ x3,     // [N,64]
    const float* __restrict__ cids,   // [N,6]  or nullptr (nn2)
    const float* __restrict__ x0in,   // [N,12] or nullptr (nn2)
    int din,                          // 64 (nn2) or 82 (nn3)
    const float* __restrict__ W0, const float* __restrict__ b0,
    const float* __restrict__ W1, const float* __restrict__ b1,
    const float* __restrict__ W2, const float* __restrict__ b2,
    const float* __restrict__ W3, const float* __restrict__ b3,
    float* __restrict__ out)          // [N,6]
{
    __shared__ float actA[16 * PIN];
    __shared__ float actB[16 * PH];
    __shared__ float actC[16 * PH];
    const int tid = threadIdx.x;
    const int w = tid >> 5;
    const int l = tid & 31;
    const int mbase = blockIdx.x * 16;

    // Stage concat(x3, cand_ids, x0) with zero K-padding (96 cols).
    for (int i = tid; i < 16 * 96; i += 256) {
        const int m = i / 96, k = i % 96;
        const int node = mbase + m;
        float v = 0.0f;
        if (k < 64) v = x3[node * 64 + k];
        else if (cids && k < 70) v = cids[node * 6 + (k - 64)];
        else if (x0in && k < 82) v = x0in[node * 12 + (k - 70)];
        actA[m * PIN + k] = v;
    }
    __syncthreads();

    const int Kpad0 = (din + 3) & ~3;            // 64 or 84
    layer_wmma(actA, PIN, actB, PH, W0, b0, din, Kpad0, 126, w, l);
    __syncthreads();
    layer_wmma(actB, PH, actC, PH, W1, b1, 126, 128, 126, w, l);
    __syncthreads();
    layer_wmma(actC, PH, actB, PH, W2, b2, 126, 128, 126, w, l);
    __syncthreads();

    // Final layer 126(->128) x 6, wave 0 only; write straight to global.
    if (w == 0) {
        const int hr = l & 15;
        const int koff = (l < 16) ? 0 : 2;
        const bool nok = (hr < 6);
        v8f acc = {0.f, 0.f, 0.f, 0.f, 0.f, 0.f, 0.f, 0.f};
        for (int k0 = 0; k0 < 128; k0 += 4) {
            const int ka = k0 + koff;
            v2f a, b;
            a.x = actB[hr * PH + ka];
            a.y = actB[hr * PH + ka + 1];
            b.x = (nok && ka < 126) ? W3[ka * 6 + hr] : 0.0f;
            b.y = (nok && ka + 1 < 126) ? W3[(ka + 1) * 6 + hr] : 0.0f;
            acc = __builtin_amdgcn_wmma_f32_16x16x4_f32(
                false, a, false, b, (short)0, acc, false, false);
        }
        if (nok) {
            const float bn = b3[hr];
            const int mo = (l < 16) ? 0 : 8;
#pragma unroll
            for (int r = 0; r < 8; ++r)
                out[(size_t)(mbase + r + mo) * 6 + hr] = lrelu(acc[r] + bn);
        }
    }
}

// ---------------------------------------------------------------------------
// Kernel 7: passthrough outputs (int ids cast to float).
// ---------------------------------------------------------------------------
__global__ void k_pass(const int* __restrict__ gi, const float* __restrict__ g,
                       const int* __restrict__ ci, const float* __restrict__ c,
                       float* __restrict__ out, int N)
{
    const int i = blockIdx.x * blockDim.x + threadIdx.x;
    if (i < N) {
        out[(size_t)N * 12 + i] = (float)gi[i];
        out[(size_t)N * 19 + i] = (float)ci[i];
    }
    if (i < N * 6) {
        out[(size_t)N * 13 + i] = g[i];
        out[(size_t)N * 20 + i] = c[i];
    }
}

// ---------------------------------------------------------------------------
extern "C" void kernel_launch(void* const* d_in, const int* in_sizes, int n_in,
                              void* d_out, int out_size, void* d_ws, size_t ws_size,
                              hipStream_t stream)
{
    const float* x0       = (const float*)d_in[0];
    const float* ygen     = (const float*)d_in[1];
    const float* ycand    = (const float*)d_in[2];
    const int*   ygen_id  = (const int*)d_in[3];
    const int*   ycand_id = (const int*)d_in[4];
    const float* n1W0 = (const float*)d_in[5];  const float* n1b0 = (const float*)d_in[6];
    const float* n1W1 = (const float*)d_in[7];  const float* n1b1 = (const float*)d_in[8];
    const float* n1W2 = (const float*)d_in[9];  const float* n1b2 = (const float*)d_in[10];
    const float* gWs  = (const float*)d_in[11]; const float* gbs  = (const float*)d_in[12];
    const float* gWh  = (const float*)d_in[13]; const float* gbh  = (const float*)d_in[14];
    const float* gWo  = (const float*)d_in[15]; const float* gbo  = (const float*)d_in[16];
    const float* c2Wrel  = (const float*)d_in[17]; const float* c2brel = (const float*)d_in[18];
    const float* c2Wroot = (const float*)d_in[19];
    const float* n2W0 = (const float*)d_in[20]; const float* n2b0 = (const float*)d_in[21];
    const float* n2W1 = (const float*)d_in[22]; const float* n2b1 = (const float*)d_in[23];
    const float* n2W2 = (const float*)d_in[24]; const float* n2b2 = (const float*)d_in[25];
    const float* n2W3 = (const float*)d_in[26]; const float* n2b3 = (const float*)d_in[27];
    const float* n3W0 = (const float*)d_in[28]; const float* n3b0 = (const float*)d_in[29];
    const float* n3W1 = (const float*)d_in[30]; const float* n3b1 = (const float*)d_in[31];
    const float* n3W2 = (const float*)d_in[32]; const float* n3b2 = (const float*)d_in[33];
    const float* n3W3 = (const float*)d_in[34]; const float* n3b3 = (const float*)d_in[35];

    const int N = in_sizes[0] / 12;   // 16384

    // Workspace layout (floats): ~251 per node (~16.5 MB total).
    float* ws  = (float*)d_ws;
    float* x   = ws;                       // N*12
    float* s   = x  + (size_t)N * 12;      // N*8
    float* h   = s  + (size_t)N * 8;       // N*22
    float* sq  = h  + (size_t)N * 22;      // N
    float* d2t = sq + (size_t)N;           // N*40
    int*   it  = (int*)(d2t + (size_t)N * KNN);   // N*40
    float* x2  = (float*)(it + (size_t)N * KNN);  // N*64
    float* x3  = x2 + (size_t)N * 64;             // N*64

    float* out = (float*)d_out;
    float* cand_ids = out;                 // N*6
    float* cand_p4  = out + (size_t)N * 6; // N*6

    k_encoder<<<N / 4, 256, 0, stream>>>(x0, n1W0, n1b0, n1W1, n1b1, n1W2, n1b2,
                                         gWs, gbs, gWh, gbh, x, s, h, sq);
    k_dist_topk<<<N / TK_ROWS, 256, 0, stream>>>(s, sq, d2t, it, N);
    k_gravnet_agg<<<N / 4, 256, 0, stream>>>(x, h, d2t, it, gWo, gbo, x2);
    k_graphconv<<<N / 4, 256, 0, stream>>>(x2, d2t, it, c2Wrel, c2brel, c2Wroot, x3);
    k_head_wmma<<<N / 16, 256, 0, stream>>>(x3, nullptr, nullptr, 64,
                                            n2W0, n2b0, n2W1, n2b1, n2W2, n2b2,
                                            n2W3, n2b3, cand_ids);
    k_head_wmma<<<N / 16, 256, 0, stream>>>(x3, cand_ids, x0, 82,
                                            n3W0, n3b0, n3W1, n3b1, n3W2, n3b2,
                                            n3W3, n3b3, cand_p4);
    k_pass<<<(N * 6 + 255) / 256, 256, 0, stream>>>(ygen_id, ygen, ycand_id, ycand,
                                                    out, N);
}